// RESGATv2_20985210208435
// MI455X (gfx1250) — compile-verified
//
#include <hip/hip_runtime.h>
#include <math.h>

#define N_NODES 20000
#define E_EDGES 640000
#define ETOT    (E_EDGES + N_NODES)   // 660000 with self loops
#define FDIM    256                   // H*C = 4*64 for every layer
#define HEADS   4
#define NEG_SLOPE 0.2f

typedef __attribute__((ext_vector_type(2))) float v2f;
typedef __attribute__((ext_vector_type(8))) float v8f;

// ---- monotonic float<->uint keys for atomic segment-max ----
__device__ __forceinline__ unsigned fkey(float f) {
  unsigned u = __float_as_uint(f);
  return (u & 0x80000000u) ? ~u : (u | 0x80000000u);
}
__device__ __forceinline__ float funkey(unsigned k) {
  return (k & 0x80000000u) ? __uint_as_float(k & 0x7FFFFFFFu)
                           : __uint_as_float(~k);
}

// ---- init: zero acc[N*F], denom[N*H]; mkey = key(-inf) ----
__global__ void k_init(float* __restrict__ acc, unsigned* __restrict__ mkey,
                       float* __restrict__ denom) {
  int i = blockIdx.x * blockDim.x + threadIdx.x;
  if (i < N_NODES * FDIM) acc[i] = 0.0f;
  if (i < N_NODES * HEADS) { mkey[i] = fkey(-INFINITY); denom[i] = 0.0f; }
}

// ---- Y[N,256] = X[N,256] @ W[256,256] + bias, fp32 WMMA 16x16x4 ----
// One wave per 16x64 output strip: A fragment loaded once per k-step and
// reused across 4 column tiles (4 WMMAs per 2 A-loads + 8 B-loads).
// A layout (32-bit 16x4): lanes0-15 K={0,1}, lanes16-31 K={2,3}.
// B layout (4x16): vgpr v = row (v + half*2), cols across lanes.
// C/D: vgpr r = row (r + half*8), cols across lanes.
__global__ void __launch_bounds__(256) k_gemm_bias(
    const float* __restrict__ X, const float* __restrict__ W,
    const float* __restrict__ bias, float* __restrict__ Y) {
  const int wave = threadIdx.x >> 5;
  const int tile = blockIdx.x * 8 + wave;              // 1250 * 4 strips
  if (tile >= (N_NODES / 16) * (FDIM / 64)) return;    // uniform per wave
  const int tileM = tile >> 2;
  const int strip = tile & 3;
  const int lane = threadIdx.x & 31;
  const int half = lane >> 4;
  const int lrow = lane & 15;
  const int row  = tileM * 16 + lrow;
  const int col0 = strip * 64 + lrow;

  const float* ap = X + (size_t)row * FDIM + 2 * half;    // A[row][k+2h+{0,1}]
  const float* bp = W + (size_t)(2 * half) * FDIM + col0; // B[k+2h+{0,1}][col0+16j]

  v8f c0 = {}, c1 = {}, c2 = {}, c3 = {};
#pragma unroll 4
  for (int k = 0; k < FDIM; k += 4) {
    v2f a;
    a.x = ap[k];
    a.y = ap[k + 1];
    const float* bk0 = bp + (size_t)k * FDIM;
    const float* bk1 = bp + (size_t)(k + 1) * FDIM;
    v2f b0, b1, b2, b3;
    b0.x = bk0[0];  b0.y = bk1[0];
    b1.x = bk0[16]; b1.y = bk1[16];
    b2.x = bk0[32]; b2.y = bk1[32];
    b3.x = bk0[48]; b3.y = bk1[48];
    c0 = __builtin_amdgcn_wmma_f32_16x16x4_f32(false, a, false, b0, (short)0, c0, false, false);
    c1 = __builtin_amdgcn_wmma_f32_16x16x4_f32(false, a, false, b1, (short)0, c1, false, false);
    c2 = __builtin_amdgcn_wmma_f32_16x16x4_f32(false, a, false, b2, (short)0, c2, false, false);
    c3 = __builtin_amdgcn_wmma_f32_16x16x4_f32(false, a, false, b3, (short)0, c3, false, false);
  }
  const float bv0 = bias[col0];
  const float bv1 = bias[col0 + 16];
  const float bv2 = bias[col0 + 32];
  const float bv3 = bias[col0 + 48];
  float* yp = Y + (size_t)(tileM * 16 + half * 8) * FDIM + col0;
#pragma unroll
  for (int r = 0; r < 8; ++r) {
    yp[(size_t)r * FDIM + 0]  = c0[r] + bv0;
    yp[(size_t)r * FDIM + 16] = c1[r] + bv1;
    yp[(size_t)r * FDIM + 32] = c2[r] + bv2;
    yp[(size_t)r * FDIM + 48] = c3[r] + bv3;
  }
}

// ---- edge scores: e[e][h] = sum_c att[h][c]*lrelu(xl[src][h,c]+xr[dst][h,c])
// one wave per edge; lane covers 8 channels (8 lanes per head) ----
__global__ void __launch_bounds__(256) k_edge_scores(
    const float* __restrict__ xl, const float* __restrict__ xr,
    const float* __restrict__ att, const int* __restrict__ eidx,
    float* __restrict__ esc, unsigned* __restrict__ mkey) {
  const int e = blockIdx.x * 8 + (threadIdx.x >> 5);
  if (e >= ETOT) return;                        // uniform per wave
  const int lane = threadIdx.x & 31;
  const int s = (e < E_EDGES) ? eidx[e] : (e - E_EDGES);
  const int d = (e < E_EDGES) ? eidx[E_EDGES + e] : (e - E_EDGES);
  const int ch0 = lane * 8;
  const float4* pl = (const float4*)(xl + (size_t)s * FDIM + ch0);
  const float4* pr = (const float4*)(xr + (size_t)d * FDIM + ch0);
  const float4* pa = (const float4*)(att + ch0);
  float part = 0.0f;
#pragma unroll
  for (int j = 0; j < 2; ++j) {
    float4 a = pl[j], b = pr[j], w = pa[j];
    float v;
    v = a.x + b.x; v = (v > 0.0f) ? v : NEG_SLOPE * v; part += w.x * v;
    v = a.y + b.y; v = (v > 0.0f) ? v : NEG_SLOPE * v; part += w.y * v;
    v = a.z + b.z; v = (v > 0.0f) ? v : NEG_SLOPE * v; part += w.z * v;
    v = a.w + b.w; v = (v > 0.0f) ? v : NEG_SLOPE * v; part += w.w * v;
  }
  // reduce within each 8-lane (per-head) group
  part += __shfl_xor(part, 1, 32);
  part += __shfl_xor(part, 2, 32);
  part += __shfl_xor(part, 4, 32);
  if ((lane & 7) == 0) {
    const int h = lane >> 3;
    esc[(size_t)e * HEADS + h] = part;
    atomicMax(&mkey[d * HEADS + h], fkey(part));
  }
}

// ---- ex = exp(e - m[dst]); denom[dst] += ex ; one thread per (edge,head) ----
__global__ void k_edge_expsum(float* __restrict__ esc,
                              const unsigned* __restrict__ mkey,
                              float* __restrict__ denom,
                              const int* __restrict__ eidx) {
  const int i = blockIdx.x * blockDim.x + threadIdx.x;
  if (i >= ETOT * HEADS) return;
  const int e = i >> 2, h = i & 3;
  const int d = (e < E_EDGES) ? eidx[E_EDGES + e] : (e - E_EDGES);
  const float m = funkey(mkey[d * HEADS + h]);
  const float ex = expf(esc[i] - m);
  esc[i] = ex;                       // overwrite score with exp in place
  atomicAdd(&denom[d * HEADS + h], ex);
}

// ---- acc[dst] += (ex/denom[dst]) * xl[src] ; one wave per edge ----
__global__ void __launch_bounds__(256) k_edge_scatter(
    const float* __restrict__ ex, const float* __restrict__ denom,
    const float* __restrict__ xl, const int* __restrict__ eidx,
    float* __restrict__ acc) {
  const int e = blockIdx.x * 8 + (threadIdx.x >> 5);
  if (e >= ETOT) return;
  const int lane = threadIdx.x & 31;
  const int s = (e < E_EDGES) ? eidx[e] : (e - E_EDGES);
  const int d = (e < E_EDGES) ? eidx[E_EDGES + e] : (e - E_EDGES);
  const int h = lane >> 3;
  const float alpha = ex[(size_t)e * HEADS + h] / denom[d * HEADS + h];
  const int ch0 = lane * 8;
  const float4* pl = (const float4*)(xl + (size_t)s * FDIM + ch0);
  float* pacc = acc + (size_t)d * FDIM + ch0;
#pragma unroll
  for (int j = 0; j < 2; ++j) {
    float4 a = pl[j];
    atomicAdd(pacc + 4 * j + 0, alpha * a.x);
    atomicAdd(pacc + 4 * j + 1, alpha * a.y);
    atomicAdd(pacc + 4 * j + 2, alpha * a.z);
    atomicAdd(pacc + 4 * j + 3, alpha * a.w);
  }
}

// ---- out = elu(acc + bo) ----
__global__ void k_epilogue_elu(const float* __restrict__ acc,
                               const float* __restrict__ bo,
                               float* __restrict__ out) {
  const int i = blockIdx.x * blockDim.x + threadIdx.x;
  if (i >= N_NODES * FDIM) return;
  const float v = acc[i] + bo[i & (FDIM - 1)];
  out[i] = (v > 0.0f) ? v : expm1f(v);
}

// ---- out = relu(elu(acc + bo3) + x_res) ----
__global__ void k_final(const float* __restrict__ acc,
                        const float* __restrict__ bo,
                        const float* __restrict__ xres,
                        float* __restrict__ out) {
  const int i = blockIdx.x * blockDim.x + threadIdx.x;
  if (i >= N_NODES * FDIM) return;
  float v = acc[i] + bo[i & (FDIM - 1)];
  v = (v > 0.0f) ? v : expm1f(v);
  v += xres[i];
  out[i] = (v > 0.0f) ? v : 0.0f;
}

extern "C" void kernel_launch(void* const* d_in, const int* in_sizes, int n_in,
                              void* d_out, int out_size, void* d_ws, size_t ws_size,
                              hipStream_t stream) {
  const float* x    = (const float*)d_in[0];
  const int*  eidx  = (const int*)d_in[1];
  const float* Wl[3] = {(const float*)d_in[2],  (const float*)d_in[8],  (const float*)d_in[14]};
  const float* bl[3] = {(const float*)d_in[3],  (const float*)d_in[9],  (const float*)d_in[15]};
  const float* Wr[3] = {(const float*)d_in[4],  (const float*)d_in[10], (const float*)d_in[16]};
  const float* br[3] = {(const float*)d_in[5],  (const float*)d_in[11], (const float*)d_in[17]};
  const float* av[3] = {(const float*)d_in[6],  (const float*)d_in[12], (const float*)d_in[18]};
  const float* bo[3] = {(const float*)d_in[7],  (const float*)d_in[13], (const float*)d_in[19]};
  const float* Wres = (const float*)d_in[20];
  const float* bres = (const float*)d_in[21];
  float* out = (float*)d_out;

  // workspace layout
  const size_t NF = (size_t)N_NODES * FDIM;
  float*    xl    = (float*)d_ws;
  float*    xr    = xl + NF;
  float*    acc   = xr + NF;
  float*    hbuf  = acc + NF;
  float*    esc   = hbuf + NF;                       // ETOT*HEADS
  unsigned* mkey  = (unsigned*)(esc + (size_t)ETOT * HEADS);
  float*    denom = (float*)(mkey + (size_t)N_NODES * HEADS);

  const int gemmBlocks  = (N_NODES / 16) * (FDIM / 64) / 8;      // 625
  const int edgeBlocks  = (ETOT + 7) / 8;                        // 82500 (wave/edge)
  const int ehBlocks    = (ETOT * HEADS + 255) / 256;
  const int nodeBlocks  = (int)((NF + 255) / 256);

  const float* layer_in = x;
  for (int L = 0; L < 3; ++L) {
    k_init<<<nodeBlocks, 256, 0, stream>>>(acc, mkey, denom);
    k_gemm_bias<<<gemmBlocks, 256, 0, stream>>>(layer_in, Wl[L], bl[L], xl);
    k_gemm_bias<<<gemmBlocks, 256, 0, stream>>>(layer_in, Wr[L], br[L], xr);
    k_edge_scores<<<edgeBlocks, 256, 0, stream>>>(xl, xr, av[L], eidx, esc, mkey);
    k_edge_expsum<<<ehBlocks, 256, 0, stream>>>(esc, mkey, denom, eidx);
    k_edge_scatter<<<edgeBlocks, 256, 0, stream>>>(esc, denom, xl, eidx, acc);
    if (L < 2) {
      k_epilogue_elu<<<nodeBlocks, 256, 0, stream>>>(acc, bo[L], hbuf);
      layer_in = hbuf;
    }
  }
  // residual branch: x_res = x @ Wres + bres (xl is free now)
  k_gemm_bias<<<gemmBlocks, 256, 0, stream>>>(x, Wres, bres, xl);
  k_final<<<nodeBlocks, 256, 0, stream>>>(acc, bo[2], xl, out);
}